// voxelization_41360535061136
// MI455X (gfx1250) — compile-verified
//
#include <hip/hip_runtime.h>
#include <hip/hip_bf16.h>

// ---------------------------------------------------------------------------
// Voxelization for MI455X (gfx1250). Memory-bound (~0.5 GB -> now ~0.3 GB
// traffic after bitmap rework; ~15us floor @ 23.3 TB/s). No GEMM structure ->
// no WMMA. CDNA5-specific paths: Tensor Data Mover staging (tensor_load_to_lds
// + s_wait_tensorcnt), LDS scans, popcount rank/select kept resident in the
// 192MB L2, wave32 sizing.
// ---------------------------------------------------------------------------

#if defined(__has_builtin)
#if __has_builtin(__builtin_amdgcn_tensor_load_to_lds) && __has_builtin(__builtin_amdgcn_s_wait_tensorcnt)
#define VOX_HAVE_TDM 1
#endif
#endif

typedef unsigned int u32x4 __attribute__((ext_vector_type(4)));
typedef int          i32x8 __attribute__((ext_vector_type(8)));
typedef int          i32x4 __attribute__((ext_vector_type(4)));

#define TILE_PTS   512                 // points staged per block via TDM
#define SCAN_B     256
#define SCAN_I     16
#define SCAN_CHUNK (SCAN_B * SCAN_I)   // 4096 elements per scan block
#define PI_F       3.14159265358979323846f

// ---------------------------------------------------------------------------
// cursor / counter control block
// ---------------------------------------------------------------------------
__global__ void init_ctrl_kernel(unsigned* ctrl, unsigned n) {
    if (threadIdx.x == 0 && blockIdx.x == 0) {
        ctrl[0] = 3u * n;  // cursor: xyz_pol occupies [0, 3N)
        ctrl[1] = 0u;      // U (unique count)
        ctrl[2] = 0u;      // M (unmask count)
    }
}

__global__ void bump_kernel(unsigned* ctrl, unsigned fixedAdd, unsigned mulU, unsigned mulM) {
    if (threadIdx.x == 0 && blockIdx.x == 0) {
        ctrl[0] += fixedAdd + mulU * ctrl[1] + mulM * ctrl[2];
    }
}

__global__ void zero_kernel(unsigned* p, unsigned n) {
    unsigned i = blockIdx.x * blockDim.x + threadIdx.x;
    if (i < n) p[i] = 0u;
}

// ---------------------------------------------------------------------------
// Kernel 1: cart2polar. Stage point tile into LDS via Tensor Data Mover,
// wait TENSORcnt, then compute rho/phi/z.
// ---------------------------------------------------------------------------
__global__ __launch_bounds__(256) void polar_kernel(const float* __restrict__ pts,
                                                    float* __restrict__ out,
                                                    int n, int totalDwords) {
    __shared__ float lds[TILE_PTS * 4];
    const int base = blockIdx.x * TILE_PTS;

#ifdef VOX_HAVE_TDM
    if (threadIdx.x < 32) {  // wave 0 issues one TDM op (EXEC ignored by TDM)
        unsigned lds_off = (unsigned)(unsigned long long)(void*)&lds[0];
        unsigned long long ga = (unsigned long long)(const void*)(pts + (size_t)base * 4);
        int remaining = totalDwords - base * 4;
        unsigned tdim0 = (remaining >= TILE_PTS * 4) ? (unsigned)(TILE_PTS * 4)
                                                     : (unsigned)(remaining > 0 ? remaining : 0);
        const unsigned tile0 = TILE_PTS * 4;   // 2048 dwords

        // D# group 0: count=1, lds_addr, global_addr[56:0], type=2
        u32x4 g0 = { 1u,
                     lds_off,
                     (unsigned)(ga & 0xFFFFFFFFull),
                     (unsigned)((ga >> 32) & 0x01FFFFFFull) | (2u << 30) };
        // D# group 1: data_size=4B (code 2); 1-D tensor: tensor_dim0=tdim0
        // (OOB tail reads return zero), tensor_dim1=1, tile_dim0=tile0.
        i32x8 g1 = { (int)(2u << 16),
                     (int)((tdim0 & 0xFFFFu) << 16),
                     (int)(((tdim0 >> 16) & 0xFFFFu) | (1u << 16)),
                     (int)(tile0 << 16),
                     0,
                     (int)tile0,
                     0, 0 };
        i32x4 g2 = { 0, 0, 0, 0 };
        i32x4 g3 = { 0, 0, 0, 0 };
        i32x8 g4 = { 0, 0, 0, 0, 0, 0, 0, 0 };
        __builtin_amdgcn_tensor_load_to_lds(g0, g1, g2, g3, g4, 0);
        __builtin_amdgcn_s_wait_tensorcnt(0);
    }
    __syncthreads();
#else
    for (int k = 0; k < (TILE_PTS * 4) / 256; ++k) {
        int d = threadIdx.x + k * 256;
        int gidx = base * 4 + d;
        lds[d] = (gidx < totalDwords) ? pts[gidx] : 0.0f;
    }
    __syncthreads();
#endif

    for (int k = 0; k < TILE_PTS / 256; ++k) {
        int j = threadIdx.x + k * 256;
        int i = base + j;
        if (i < n) {
            float x = lds[4 * j + 0];
            float y = lds[4 * j + 1];
            float z = lds[4 * j + 2];
            out[3 * i + 0] = sqrtf(x * x + y * y);
            out[3 * i + 1] = atan2f(y, x);
            out[3 * i + 2] = z;
        }
    }
}

// ---------------------------------------------------------------------------
// Kernel 2 (per scale): quantize, write full_coors + keys, set occupancy bit.
// atomicOr result is order-independent -> deterministic across replays.
// ---------------------------------------------------------------------------
__global__ void grid_kernel(const float* __restrict__ pol,
                            const int* __restrict__ batch,
                            unsigned* __restrict__ keys,
                            unsigned* __restrict__ bitmap,
                            float* __restrict__ out,
                            const unsigned* __restrict__ ctrl,
                            int n, int Sx, int Sy, int Sz) {
    int i = blockIdx.x * blockDim.x + threadIdx.x;
    if (i >= n) return;
    unsigned cur = ctrl[0];

    float r = pol[3 * i + 0];
    float p = pol[3 * i + 1];
    float z = pol[3 * i + 2];
    r = fminf(fmaxf(r, 0.0f), 50.0f);
    p = fminf(fmaxf(p, -PI_F), PI_F);
    z = fminf(fmaxf(z, -4.0f), 2.0f);

    int gx = (int)floorf(r * (1.0f / 50.0f) * (float)(Sx - 1));
    int gy = (int)floorf((p + PI_F) * (1.0f / (2.0f * PI_F)) * (float)(Sy - 1));
    int gz = (int)floorf((z + 4.0f) * (1.0f / 6.0f) * (float)(Sz - 1));
    gx = min(max(gx, 0), Sx - 1);
    gy = min(max(gy, 0), Sy - 1);
    gz = min(max(gz, 0), Sz - 1);

    int b = batch[i];
    unsigned key = ((unsigned)((b * Sx + gx) * Sy + gy)) * (unsigned)Sz + (unsigned)gz;
    keys[i] = key;
    atomicOr(&bitmap[key >> 5], 1u << (key & 31u));

    unsigned o = cur + 4u * (unsigned)i;
    out[o + 0] = (float)b;
    out[o + 1] = (float)gx;
    out[o + 2] = (float)gy;
    out[o + 3] = (float)gz;
}

// ---------------------------------------------------------------------------
// 3-phase exclusive scan. pop=1: element value = popcount(word) (bitmap scan).
// pop=0: element value = (word != 0) (flag scan).
// ---------------------------------------------------------------------------
__device__ __forceinline__ unsigned elem_val(unsigned v, int pop) {
    return pop ? (unsigned)__popc(v) : (v != 0u ? 1u : 0u);
}

__global__ __launch_bounds__(SCAN_B) void scanA_kernel(const unsigned* __restrict__ data,
                                                       unsigned* __restrict__ bsums,
                                                       const unsigned* __restrict__ nptr,
                                                       unsigned nfix, int pop) {
    __shared__ unsigned s[SCAN_B];
    unsigned n = nptr ? nptr[0] : nfix;
    unsigned base = (unsigned)blockIdx.x * SCAN_CHUNK;
    unsigned sum = 0;
    if (base < n) {
        unsigned start = base + threadIdx.x * SCAN_I;  // contiguous per thread
        for (int k = 0; k < SCAN_I; ++k) {
            unsigned idx = start + k;
            if (idx < n) sum += elem_val(data[idx], pop);
        }
    }
    s[threadIdx.x] = sum;
    __syncthreads();
    for (int off = SCAN_B / 2; off > 0; off >>= 1) {
        if ((int)threadIdx.x < off) s[threadIdx.x] += s[threadIdx.x + off];
        __syncthreads();
    }
    if (threadIdx.x == 0) bsums[blockIdx.x] = s[0];
}

// Single-workgroup looping LDS scan over block sums (parallel, not serial).
__global__ __launch_bounds__(SCAN_B) void scanB_kernel(unsigned* __restrict__ bsums,
                                                       const unsigned* __restrict__ nptr,
                                                       unsigned nfix, unsigned* totalDst) {
    __shared__ unsigned s[SCAN_B];
    unsigned n = nptr ? nptr[0] : nfix;
    unsigned nb = (n + SCAN_CHUNK - 1) / SCAN_CHUNK;
    unsigned carry = 0;
    for (unsigned b0 = 0; b0 < nb; b0 += SCAN_B) {
        unsigned i = b0 + threadIdx.x;
        unsigned v = (i < nb) ? bsums[i] : 0u;
        s[threadIdx.x] = v;
        __syncthreads();
        for (int off = 1; off < SCAN_B; off <<= 1) {
            unsigned t = (threadIdx.x >= (unsigned)off) ? s[threadIdx.x - off] : 0u;
            __syncthreads();
            s[threadIdx.x] += t;
            __syncthreads();
        }
        unsigned exc = (threadIdx.x == 0) ? 0u : s[threadIdx.x - 1];
        if (i < nb) bsums[i] = carry + exc;
        unsigned blockTotal = s[SCAN_B - 1];
        __syncthreads();
        carry += blockTotal;
    }
    if (threadIdx.x == 0 && totalDst) *totalDst = carry;
}

// mode 0 (bitmap): aux[idx] = exclusive popcount prefix at word granularity.
// mode 1 (flags):  compact flagged indices: out[ctrl[0] + rank] = (float)idx.
__global__ __launch_bounds__(SCAN_B) void scanC_kernel(const unsigned* __restrict__ data,
                                                       const unsigned* __restrict__ bsums,
                                                       const unsigned* __restrict__ nptr,
                                                       unsigned nfix, int mode,
                                                       unsigned* __restrict__ aux,
                                                       float* __restrict__ out,
                                                       const unsigned* __restrict__ ctrl) {
    __shared__ unsigned s[SCAN_B];
    unsigned n = nptr ? nptr[0] : nfix;
    unsigned base = (unsigned)blockIdx.x * SCAN_CHUNK;
    if (base >= n) return;  // uniform per block

    const int pop = (mode == 0) ? 1 : 0;
    unsigned cur = (mode == 1) ? ctrl[0] : 0u;
    unsigned start = base + threadIdx.x * SCAN_I;
    unsigned mysum = 0;
    for (int k = 0; k < SCAN_I; ++k) {
        unsigned idx = start + k;
        if (idx < n) mysum += elem_val(data[idx], pop);
    }
    s[threadIdx.x] = mysum;
    __syncthreads();
    for (int off = 1; off < SCAN_B; off <<= 1) {
        unsigned v = (threadIdx.x >= (unsigned)off) ? s[threadIdx.x - off] : 0u;
        __syncthreads();
        s[threadIdx.x] += v;
        __syncthreads();
    }
    unsigned exc = (threadIdx.x == 0) ? 0u : s[threadIdx.x - 1];
    unsigned run = bsums[blockIdx.x] + exc;
    for (int k = 0; k < SCAN_I; ++k) {
        unsigned idx = start + k;
        if (idx < n) {
            unsigned v = data[idx];
            if (mode == 0) {
                aux[idx] = run;
                run += (unsigned)__popc(v);
            } else if (v != 0u) {
                out[cur + run] = (float)idx;
                run++;
            }
        }
    }
}

// ---------------------------------------------------------------------------
// unq_inv: rank(key) = wordpfx[key>>5] + popc(bitmap_word & below-mask).
// bitmap+wordpfx (~11MB at scale 1) stay resident in the 192MB L2.
// ---------------------------------------------------------------------------
__global__ void inv_kernel(const unsigned* __restrict__ keys,
                           const unsigned* __restrict__ bitmap,
                           const unsigned* __restrict__ wordpfx,
                           float* __restrict__ out,
                           const unsigned* __restrict__ ctrl, int n) {
    int i = blockIdx.x * blockDim.x + threadIdx.x;
    if (i >= n) return;
    __builtin_prefetch(&keys[i + 8192], 0, 0);  // global_prefetch_b8
    unsigned cur = ctrl[0];
    unsigned key = keys[i];
    unsigned w = key >> 5, b = key & 31u;
    unsigned below = bitmap[w] & ((b == 0u) ? 0u : (0xFFFFFFFFu >> (32u - b)));
    out[cur + (unsigned)i] = (float)(wordpfx[w] + (unsigned)__popc(below));
}

// ---------------------------------------------------------------------------
// coors decode (sorted-unique order) + deterministic per-band keep flags.
// One thread per 32-key bitmap word; set bits iterated in ascending order.
// cursor currently at inv slot -> coors base = cur + N.
// ---------------------------------------------------------------------------
__global__ void coors_kernel(const unsigned* __restrict__ bitmap,
                             const unsigned* __restrict__ wordpfx,
                             unsigned* __restrict__ flags,
                             float* __restrict__ out,
                             const unsigned* __restrict__ ctrl,
                             unsigned wn, int n, int Sx, int Sy, int Sz,
                             float voxel_r) {
    unsigned w = blockIdx.x * blockDim.x + threadIdx.x;
    if (w >= wn) return;
    unsigned word = bitmap[w];
    if (word == 0u) return;
    unsigned rank = wordpfx[w];
    unsigned cur = ctrl[0];

    while (word) {
        unsigned bit = (unsigned)__ffs(word) - 1u;
        word &= word - 1u;
        unsigned key = (w << 5) + bit;

        unsigned gz = key % (unsigned)Sz;
        unsigned t = key / (unsigned)Sz;
        unsigned gy = t % (unsigned)Sy;
        t /= (unsigned)Sy;
        unsigned gx = t % (unsigned)Sx;
        unsigned b = t / (unsigned)Sx;

        unsigned o = cur + (unsigned)n + 4u * rank;
        out[o + 0] = (float)b;
        out[o + 1] = (float)gz;
        out[o + 2] = (float)gy;
        out[o + 3] = (float)gx;

        // mask-ratio bands on radial voxel center (reference: 0.9 / 0.7 / 0.5)
        float center = ((float)gx + 0.5f) * voxel_r;
        float ratio = (center <= 20.0f) ? 0.9f : ((center <= 40.0f) ? 0.7f : 0.5f);
        unsigned h = rank * 2654435761u;
        h ^= h >> 16;
        flags[rank] = (((float)(h & 0xFFFFu) * (1.0f / 65536.0f)) >= ratio) ? 1u : 0u;
        rank++;
    }
}

// ---------------------------------------------------------------------------
// Host launcher
// ---------------------------------------------------------------------------
extern "C" void kernel_launch(void* const* d_in, const int* in_sizes, int n_in,
                              void* d_out, int out_size, void* d_ws, size_t ws_size,
                              hipStream_t stream) {
    const float* points = (const float*)d_in[0];
    const int* batch = (const int*)d_in[1];
    const int n = in_sizes[1];  // N_POINTS (points is N x 4)
    float* out = (float*)d_out;

    const int SXs[5] = {240, 120, 60, 30, 480};  // ceil(480/scale)
    const int SYs[5] = {180,  90, 45, 23, 360};  // ceil(360/scale)
    const int SZs[5] = { 16,   8,  4,  2,  32};  // ceil(32/scale)
    const int BATCH = 8;

    // max bitmap words (scale 1: 8*480*360*32 / 32 keys per word)
    const size_t WMAX = ((size_t)BATCH * 480 * 360 * 32 + 31) / 32;  // 1,382,400

    // workspace carve-out (~19.3 MB total)
    unsigned char* ws = (unsigned char*)d_ws;
    unsigned* ctrl    = (unsigned*)ws;                                  // 64 u32
    unsigned* bsums   = (unsigned*)(ws + 256);                          // 16384 u32
    unsigned* keys    = (unsigned*)(ws + 256 + 16384 * 4);              // N u32 (reused as flags)
    unsigned* bitmap  = (unsigned*)(ws + 256 + 16384 * 4 + (size_t)n * 4);
    unsigned* wordpfx = bitmap + WMAX;

    const int TPB = 256;
    const int nbPts = (n + TPB - 1) / TPB;
    const int nbTile = (n + TILE_PTS - 1) / TILE_PTS;
    const int nbScanN = (n + SCAN_CHUNK - 1) / SCAN_CHUNK;

    init_ctrl_kernel<<<1, 1, 0, stream>>>(ctrl, (unsigned)n);
    polar_kernel<<<nbTile, TPB, 0, stream>>>(points, out, n, n * 4);

    for (int sc = 0; sc < 5; ++sc) {
        const int Sx = SXs[sc], Sy = SYs[sc], Sz = SZs[sc];
        const unsigned dn = (unsigned)(BATCH * Sx * Sy * Sz);
        const unsigned wn = (dn + 31u) / 32u;
        const int nbWords = (int)((wn + TPB - 1) / TPB);
        const int nbScanW = (int)((wn + SCAN_CHUNK - 1) / SCAN_CHUNK);

        // 1) occupancy bitmap + keys + full_coors (cursor at fc slot)
        zero_kernel<<<nbWords, TPB, 0, stream>>>(bitmap, wn);
        grid_kernel<<<nbPts, TPB, 0, stream>>>(out, batch, keys, bitmap, out, ctrl,
                                               n, Sx, Sy, Sz);
        bump_kernel<<<1, 1, 0, stream>>>(ctrl, 4u * (unsigned)n, 0u, 0u);  // -> inv slot

        // 2) popcount scan over bitmap -> per-word exclusive prefix, U total
        scanA_kernel<<<nbScanW, SCAN_B, 0, stream>>>(bitmap, bsums, nullptr, wn, 1);
        scanB_kernel<<<1, SCAN_B, 0, stream>>>(bsums, nullptr, wn, ctrl + 1);   // U
        scanC_kernel<<<nbScanW, SCAN_B, 0, stream>>>(bitmap, bsums, nullptr, wn,
                                                     0, wordpfx, out, ctrl);

        // 3) unq_inv then coors (+ keep flags into keys region)
        inv_kernel<<<nbPts, TPB, 0, stream>>>(keys, bitmap, wordpfx, out, ctrl, n);
        coors_kernel<<<nbWords, TPB, 0, stream>>>(bitmap, wordpfx, keys, out, ctrl,
                                                  wn, n, Sx, Sy, Sz, 50.0f / (float)Sx);
        bump_kernel<<<1, 1, 0, stream>>>(ctrl, (unsigned)n, 4u, 0u);       // -> unmask slot

        // 4) compact kept unique ids (ascending) = unmask_index
        scanA_kernel<<<nbScanN, SCAN_B, 0, stream>>>(keys, bsums, ctrl + 1, 0u, 0);
        scanB_kernel<<<1, SCAN_B, 0, stream>>>(bsums, ctrl + 1, 0u, ctrl + 2);  // M
        scanC_kernel<<<nbScanN, SCAN_B, 0, stream>>>(keys, bsums, ctrl + 1, 0u,
                                                     1, nullptr, out, ctrl);
        bump_kernel<<<1, 1, 0, stream>>>(ctrl, 0u, 0u, 1u);                // -> next fc slot
    }
}